// CoarseMatching_54400055771233
// MI455X (gfx1250) — compile-verified
//
#include <hip/hip_runtime.h>
#include <hip/hip_bf16.h>
#include <cstdint>
#include <cstddef>

#define THR       0.2f
#define BORDER    2
#define K1_CHUNK  128    // rows per colmax block
#define K2_WAVES  8      // rows (waves) per match block
#define SMAX      4800   // max S staged in LDS (19.2 KB of the 320 KB WGP LDS)
#define NEG_HUGE  -3.402823466e38f

// ---- CDNA5 async global->LDS feature detection -----------------------------
#if defined(__has_builtin)
#  if __has_builtin(__builtin_amdgcn_global_load_async_to_lds_b128)
#    define HAVE_ASYNC_LDS 1
#  endif
#  if __has_builtin(__builtin_amdgcn_s_wait_asynccnt)
#    define HAVE_WAIT_ASYNC 1
#  endif
#endif
#ifndef HAVE_ASYNC_LDS
#define HAVE_ASYNC_LDS 0
#endif
#ifndef HAVE_WAIT_ASYNC
#define HAVE_WAIT_ASYNC 0
#endif

#if HAVE_ASYNC_LDS
// Matches the builtin's parameter type reported by clang:
// '__attribute__((__vector_size__(4 * sizeof(int)))) int __device__ *'
typedef int v4i_b128 __attribute__((vector_size(4 * sizeof(int))));
typedef __attribute__((address_space(1))) v4i_b128* glb_v4i_ptr;   // __device__
typedef __attribute__((address_space(3))) v4i_b128* lds_v4i_ptr;   // __shared__

__device__ __forceinline__ void async_copy_b128(const unsigned* gsrc, unsigned* ldst) {
  __builtin_amdgcn_global_load_async_to_lds_b128(
      (glb_v4i_ptr)(gsrc), (lds_v4i_ptr)(ldst), 0, 0);
}
#endif

// Order-preserving float->u32 key: unsigned compare == float compare,
// and key equality == float equality (injective). Lets colmax live as u32
// so GLOBAL_ATOMIC_MAX_U32 computes the float column max.
__device__ __forceinline__ unsigned enc_key(float f) {
  unsigned b = __float_as_uint(f);
  return b ^ ((unsigned)((int)b >> 31) | 0x80000000u);
}

// ---------------------------------------------------------------------------
// Per-lane fused row scan: running rowmax + earliest index of the running
// candidate max among entries passing (thr & ==colmax-key). Strict '>' update
// keeps the first occurrence; lane j-sequence is monotone.
// ---------------------------------------------------------------------------
struct ScanRes { float M; float cmax; int cj; };

__device__ __forceinline__ ScanRes scan_row(const float* __restrict__ row,
                                            const unsigned* __restrict__ cm,
                                            int S, int lane) {
  ScanRes r; r.M = NEG_HUGE; r.cmax = NEG_HUGE; r.cj = 0x7fffffff;
  if ((S & 3) == 0) {
    const float4* r4 = (const float4*)row;
    const uint4*  c4 = (const uint4*)cm;
    const int n4 = S >> 2;
    for (int q = lane; q < n4; q += 32) {
      const float4 v = r4[q];
      const uint4  c = c4[q];
      const int j0 = q << 2;
      { float f = v.x; r.M = fmaxf(r.M, f);
        if (f > THR && enc_key(f) == c.x && f > r.cmax) { r.cmax = f; r.cj = j0;     } }
      { float f = v.y; r.M = fmaxf(r.M, f);
        if (f > THR && enc_key(f) == c.y && f > r.cmax) { r.cmax = f; r.cj = j0 + 1; } }
      { float f = v.z; r.M = fmaxf(r.M, f);
        if (f > THR && enc_key(f) == c.z && f > r.cmax) { r.cmax = f; r.cj = j0 + 2; } }
      { float f = v.w; r.M = fmaxf(r.M, f);
        if (f > THR && enc_key(f) == c.w && f > r.cmax) { r.cmax = f; r.cj = j0 + 3; } }
    }
  } else {
    for (int j = lane; j < S; j += 32) {
      const float f = row[j]; r.M = fmaxf(r.M, f);
      if (f > THR && enc_key(f) == cm[j] && f > r.cmax) { r.cmax = f; r.cj = j; }
    }
  }
  return r;
}

// ---------------------------------------------------------------------------
// Kernel 2 (placed first so the disasm snippet shows the async-LDS path):
// one wave32 per row. Block stages colmax keys for its n into LDS via CDNA5
// async global->LDS (b128) + s_wait_asynccnt, then each wave does one fused
// scan and wave32 butterfly reductions.
// ---------------------------------------------------------------------------
__global__ __launch_bounds__(256) void k_match(
    const float* __restrict__ conf, const unsigned* __restrict__ colmax,
    const int* __restrict__ ph0, const int* __restrict__ pw0,
    const int* __restrict__ ph1, const int* __restrict__ pw1,
    int NL, float* __restrict__ out) {
  const int h0 = *ph0, w0 = *pw0, h1 = *ph1, w1 = *pw1;
  const int L = h0 * w0, S = h1 * w1;

  __shared__ __align__(16) unsigned s_cm[SMAX];

  const int r_base = blockIdx.x * K2_WAVES;
  const int rb = (r_base < NL) ? r_base : (NL - 1);
  const int staged_n = rb / L;
  const bool can_lds = (S <= SMAX);

  if (can_lds) {  // all threads participate (no early returns before barrier)
    const unsigned* g = colmax + (size_t)staged_n * S;
#if HAVE_ASYNC_LDS
    if ((S & 3) == 0) {
      const int n4 = S >> 2;
      for (int q = threadIdx.x; q < n4; q += blockDim.x) {
        async_copy_b128(g + 4 * q, s_cm + 4 * q);
      }
#if HAVE_WAIT_ASYNC
      __builtin_amdgcn_s_wait_asynccnt(0);
#else
      asm volatile("s_wait_asynccnt 0" ::: "memory");
#endif
    } else {
      for (int q = threadIdx.x; q < S; q += blockDim.x) s_cm[q] = g[q];
    }
#else
    for (int q = threadIdx.x; q < S; q += blockDim.x) s_cm[q] = g[q];
#endif
    __syncthreads();
  }

  const int wid  = threadIdx.x >> 5;
  const int lane = threadIdx.x & 31;
  const int r = r_base + wid;
  if (r >= NL) return;

  const int n = r / L;
  const int i = r - n * L;
  float* out_conf = out;
  float* out_mask = out + NL;
  float* out_jid  = out + 2 * (size_t)NL;

  const int i0r = i / w0, i0c = i - i0r * w0;
  const bool valid0 = (i0r >= BORDER) && (i0r < h0 - BORDER) &&
                      (i0c >= BORDER) && (i0c < w0 - BORDER);
  if (!valid0) {
    if (lane == 0) { out_conf[r] = 0.f; out_mask[r] = 0.f; out_jid[r] = 0.f; }
    return;
  }

  const float* row = conf + (size_t)r * S;
  ScanRes sr;
  if (can_lds && n == staged_n) sr = scan_row(row, s_cm, S, lane);             // DS path
  else                          sr = scan_row(row, colmax + (size_t)n * S, S, lane);

  float M = sr.M;
#pragma unroll
  for (int o = 16; o > 0; o >>= 1) M = fmaxf(M, __shfl_xor(M, o, 32));
  int js = (sr.cj != 0x7fffffff && sr.cmax == M) ? sr.cj : 0x7fffffff;
#pragma unroll
  for (int o = 16; o > 0; o >>= 1) { int t = __shfl_xor(js, o, 32); js = (t < js) ? t : js; }

  if (lane == 0) {
    const bool hit = (js != 0x7fffffff);
    out_conf[r] = hit ? M : 0.f;          // matched conf == rowmax by construction
    out_mask[r] = hit ? 1.f : 0.f;
    out_jid[r]  = hit ? (float)js : 0.f;  // argmax(all-false) == 0 in reference
  }
}

// ---------------------------------------------------------------------------
// Kernel 1: column max (per n, per j) as encoded keys, valid1 folded in.
// Thread owns 4 consecutive columns; block.y chunks rows; atomic-max flush
// at n boundaries and chunk end. Fully coalesced float4 streaming.
// ---------------------------------------------------------------------------
__global__ __launch_bounds__(256) void k_colmax(
    const float* __restrict__ conf,
    const int* __restrict__ ph0, const int* __restrict__ pw0,
    const int* __restrict__ ph1, const int* __restrict__ pw1,
    int NL, unsigned* __restrict__ colmax) {
  const int h1 = *ph1, w1 = *pw1;
  const int S  = h1 * w1;
  const int L  = (*ph0) * (*pw0);

  const int j4 = (blockIdx.x * blockDim.x + threadIdx.x) * 4;
  if (j4 >= S) return;
  const int ne = (S - j4) < 4 ? (S - j4) : 4;

  bool val[4]; bool anyv = false;
#pragma unroll
  for (int e = 0; e < 4; ++e) {
    const int j  = j4 + e;
    const int r1 = j / w1, c1 = j - r1 * w1;
    val[e] = (e < ne) && (r1 >= BORDER) && (r1 < h1 - BORDER) &&
             (c1 >= BORDER) && (c1 < w1 - BORDER);
    anyv |= val[e];
  }
  if (!anyv) return;  // invalid columns keep key 0 -> never match in pass 2

  const int r0 = blockIdx.y * K1_CHUNK;
  if (r0 >= NL) return;
  int rend = r0 + K1_CHUNK; if (rend > NL) rend = NL;

  int n    = r0 / L;
  int i_in = r0 - n * L;
  int rows_acc = 0;
  float m0 = NEG_HUGE, m1 = NEG_HUGE, m2 = NEG_HUGE, m3 = NEG_HUGE;
  const bool vec = (ne == 4) && ((S & 3) == 0);
  const float* p = conf + (size_t)r0 * S + j4;

  for (int r = r0; r < rend; ++r, p += S) {
    if (vec) {
      const float4 v = *(const float4*)p;
      m0 = fmaxf(m0, v.x); m1 = fmaxf(m1, v.y);
      m2 = fmaxf(m2, v.z); m3 = fmaxf(m3, v.w);
    } else {
      if (ne > 0) m0 = fmaxf(m0, p[0]);
      if (ne > 1) m1 = fmaxf(m1, p[1]);
      if (ne > 2) m2 = fmaxf(m2, p[2]);
    }
    ++rows_acc;
    if (++i_in == L) {  // crossed an n boundary: flush and reset
      unsigned* dst = colmax + (size_t)n * S + j4;
      if (val[0]) atomicMax(&dst[0], enc_key(m0));
      if (val[1]) atomicMax(&dst[1], enc_key(m1));
      if (val[2]) atomicMax(&dst[2], enc_key(m2));
      if (val[3]) atomicMax(&dst[3], enc_key(m3));
      m0 = m1 = m2 = m3 = NEG_HUGE;
      rows_acc = 0; i_in = 0; ++n;
    }
  }
  if (rows_acc > 0) {
    unsigned* dst = colmax + (size_t)n * S + j4;
    if (val[0]) atomicMax(&dst[0], enc_key(m0));
    if (val[1]) atomicMax(&dst[1], enc_key(m1));
    if (val[2]) atomicMax(&dst[2], enc_key(m2));
    if (val[3]) atomicMax(&dst[3], enc_key(m3));
  }
}

// ---------------------------------------------------------------------------
// Kernel 0: zero the colmax key table (deterministic across graph replays).
// count = N*S, with N = NL/L; 32-bit scalar division only.
// ---------------------------------------------------------------------------
__global__ void k_init(unsigned* __restrict__ cm,
                       const int* __restrict__ ph0, const int* __restrict__ pw0,
                       const int* __restrict__ ph1, const int* __restrict__ pw1,
                       int NL) {
  const int L = (*ph0) * (*pw0);
  const int S = (*ph1) * (*pw1);
  const int N = NL / L;
  const size_t cnt = (size_t)N * S;
  const size_t stride = (size_t)gridDim.x * blockDim.x;
  for (size_t x = (size_t)blockIdx.x * blockDim.x + threadIdx.x; x < cnt; x += stride)
    cm[x] = 0u;
}

// ---------------------------------------------------------------------------
extern "C" void kernel_launch(void* const* d_in, const int* in_sizes, int n_in,
                              void* d_out, int out_size, void* d_ws, size_t ws_size,
                              hipStream_t stream) {
  (void)n_in; (void)ws_size;
  const float* conf = (const float*)d_in[0];
  const int* ph0 = (const int*)d_in[1];
  const int* pw0 = (const int*)d_in[2];
  const int* ph1 = (const int*)d_in[3];
  const int* pw1 = (const int*)d_in[4];

  const long long total = (long long)in_sizes[0];      // N*L*S
  const int NL = out_size / 3;                         // N*L
  const int S  = (int)((3LL * total) / (long long)out_size);
  unsigned* colmax = (unsigned*)d_ws;                  // N*S keys
  float* out = (float*)d_out;

  k_init<<<256, 256, 0, stream>>>(colmax, ph0, pw0, ph1, pw1, NL);

  const int S4 = (S + 3) / 4;
  dim3 g1((S4 + 255) / 256, (NL + K1_CHUNK - 1) / K1_CHUNK);
  k_colmax<<<g1, 256, 0, stream>>>(conf, ph0, pw0, ph1, pw1, NL, colmax);

  dim3 g2((NL + K2_WAVES - 1) / K2_WAVES);
  k_match<<<g2, 256, 0, stream>>>(conf, colmax, ph0, pw0, ph1, pw1, NL, out);
}